// CSplinesPSF_69509750719066
// MI455X (gfx1250) — compile-verified
//
#include <hip/hip_runtime.h>

// ---------------------------------------------------------------------------
// CSplinesPSF renderer for MI455X (gfx1250, wave32).
//   B=8, N=512, F=16, D=150, patch 26x26, image 128x128.
// Fused render+photon-weight GEMM per emitter via V_WMMA_F32_16X16X4_F32:
//   patch[f, p] = sum_c (photons[f]*series[c]) * coeffs[z, p, c]
//   A (16 frames x 64 coeffs) x B (64 coeffs x 16 pixels) -> D (16x16)
// Scatter-add via global_atomic_add_f32 into an L2-resident padded canvas.
// ---------------------------------------------------------------------------

typedef __attribute__((ext_vector_type(2))) float v2f;
typedef __attribute__((ext_vector_type(8))) float v8f;

#define B_    8
#define N_    512
#define F_    16
#define D_    150
#define PH_   26
#define PW_   26
#define IMG_  128
#define HP_   (IMG_ + 2 * PH_)        // 180
#define WP_   (IMG_ + 2 * PW_)        // 180
#define NPIX  (PH_ * PW_)             // 676
#define NTILE ((NPIX + 15) / 16)      // 43
#define NE    (B_ * N_)               // 4096
#define CANVAS_ELEMS (B_ * F_ * HP_ * WP_)   // 4,147,200 floats (~16.6 MB)
#define OUT_ELEMS    (B_ * F_ * IMG_ * IMG_) // 2,097,152 floats

// ---------------------------------------------------------------------------
__global__ void psf_zero_kernel(float* __restrict__ p, int n) {
    int i = blockIdx.x * blockDim.x + threadIdx.x;
    if (i < n) p[i] = 0.0f;
}

// ---------------------------------------------------------------------------
// One thread per emitter: integer indices + 64-term cubic power series.
__global__ void psf_pre_kernel(const float* __restrict__ xyz,
                               const float* __restrict__ ivs,
                               const float* __restrict__ pc,
                               float* __restrict__ series,   // (NE, 64)
                               int*   __restrict__ eidx) {   // (NE, 4)
    int e = blockIdx.x * blockDim.x + threadIdx.x;
    if (e >= NE) return;

    float ux = xyz[3 * e + 0] * ivs[0] - pc[0];
    float uy = xyz[3 * e + 1] * ivs[1] - pc[1];
    float uz = xyz[3 * e + 2] * ivs[2] + pc[2];

    float fxf = floorf(ux), fyf = floorf(uy), fzf = floorf(uz);
    int ix = (int)fxf, iy = (int)fyf, iz = (int)fzf;

    // frac flip on x/y per reference
    float fx = 1.0f - (ux - fxf);
    float fy = 1.0f - (uy - fyf);
    float fz = uz - fzf;

    int x_idx = ix + PW_;
    int y_idx = iy + PH_;
    int z_idx = iz;
    if (z_idx < 0) z_idx = 0;
    if (z_idx > D_ - 1) z_idx = D_ - 1;

    float X[4] = {1.0f, fx, fx * fx, fx * fx * fx};
    float Y[4] = {1.0f, fy, fy * fy, fy * fy * fy};
    float Z[4] = {1.0f, fz, fz * fz, fz * fz * fz};

    float* s = series + (size_t)e * 64;
    #pragma unroll
    for (int pz = 0; pz < 4; ++pz)
        #pragma unroll
        for (int px = 0; px < 4; ++px)
            #pragma unroll
            for (int py = 0; py < 4; ++py)
                s[pz * 16 + px * 4 + py] = Z[pz] * X[px] * Y[py];

    eidx[4 * e + 0] = x_idx;
    eidx[4 * e + 1] = y_idx;
    eidx[4 * e + 2] = z_idx;
    eidx[4 * e + 3] = 0;
}

// ---------------------------------------------------------------------------
// One wave32 per emitter. Fused GEMM + scatter.
// WMMA fragment convention (same K-slot map for A and B; sum over K is
// invariant under a common K permutation):
//   slot (vgpr j, lane-half h) within a K=4 chunk kc holds c = 4*kc + 2*h + j
//   A: M = f      = lane&15 ; a[j] = photons[f] * series[c]
//   B: N = pixel  = lane&15 ; b[j] = coeffs[z, p0 + (lane&15), c]
//   D (ISA layout): element (f = r + 8*half, pixel = p0 + (lane&15)) in vgpr r
__global__ __launch_bounds__(32) void psf_main_kernel(
        const float* __restrict__ coeffs,   // (D, 26, 26, 64)
        const float* __restrict__ photons,  // (B, N, F) == (NE, F)
        const float* __restrict__ series,   // (NE, 64)
        const int*   __restrict__ eidx,     // (NE, 4)
        float* __restrict__ canvas) {       // (B*F, HP, WP)
    const int e    = blockIdx.x;
    const int lane = threadIdx.x;
    const int half = lane >> 4;
    const int l15  = lane & 15;
    const int b    = e / N_;

    const int x_idx = eidx[4 * e + 0];
    const int y_idx = eidx[4 * e + 1];
    const int z_idx = eidx[4 * e + 2];

    const float photon = photons[(size_t)e * F_ + l15];

    // A fragments for all 16 K-chunks (reused across all 43 pixel tiles).
    v2f a[16];
    const v2f* s2 = (const v2f*)(series + (size_t)e * 64);
    #pragma unroll
    for (int kc = 0; kc < 16; ++kc) {
        v2f sv = s2[2 * kc + half];
        a[kc].x = photon * sv.x;
        a[kc].y = photon * sv.y;
    }

    const float* slab = coeffs + (size_t)z_idx * (NPIX * 64);

    for (int t = 0; t < NTILE; ++t) {
        const int p0 = t * 16;
        const int p  = p0 + l15;
        const bool valid = (p < NPIX);
        const int  pcl   = valid ? p : (NPIX - 1);
        const float* bp  = slab + (size_t)pcl * 64 + 2 * half;

        // Prefetch next tile's slab rows (L2-resident, warms WGP$).
        if (t + 1 < NTILE) {
            int pn = p0 + 16 + l15;
            if (pn > NPIX - 1) pn = NPIX - 1;
            __builtin_prefetch(slab + (size_t)pn * 64 + 2 * half, 0, 0);
        }

        v8f acc = {0.f, 0.f, 0.f, 0.f, 0.f, 0.f, 0.f, 0.f};
        #pragma unroll
        for (int kc = 0; kc < 16; ++kc) {
            v2f bv = *(const v2f*)(bp + 4 * kc);
            acc = __builtin_amdgcn_wmma_f32_16x16x4_f32(
                      /*neg_a=*/false, a[kc],
                      /*neg_b=*/false, bv,
                      /*c_mod=*/(short)0, acc,
                      /*reuse_a=*/false, /*reuse_b=*/false);
        }

        if (valid) {
            const int dy = p / PW_;
            const int dx = p - dy * PW_;
            const int rowoff = (y_idx + dy) * WP_ + (x_idx + dx);
            #pragma unroll
            for (int r = 0; r < 8; ++r) {
                const int f = r + 8 * half;
                atomicAdd(canvas + ((size_t)(b * F_ + f) * (HP_ * WP_) + rowoff),
                          acc[r]);
            }
        }
    }
}

// ---------------------------------------------------------------------------
__global__ void psf_crop_kernel(const float* __restrict__ canvas,
                                float* __restrict__ out) {
    int i = blockIdx.x * blockDim.x + threadIdx.x;
    if (i >= OUT_ELEMS) return;
    int x  = i & (IMG_ - 1);
    int y  = (i >> 7) & (IMG_ - 1);
    int bf = i >> 14;
    out[i] = canvas[((size_t)bf * HP_ + (y + PH_)) * WP_ + (x + PW_)];
}

// ---------------------------------------------------------------------------
extern "C" void kernel_launch(void* const* d_in, const int* in_sizes, int n_in,
                              void* d_out, int out_size, void* d_ws, size_t ws_size,
                              hipStream_t stream) {
    const float* xyz    = (const float*)d_in[0];  // (B,N,3)
    const float* nph    = (const float*)d_in[1];  // (B,N,F)
    const float* coeffs = (const float*)d_in[2];  // (D,26,26,64)
    const float* ivs    = (const float*)d_in[3];  // (3,)
    const float* pc     = (const float*)d_in[4];  // (3,)
    (void)in_sizes; (void)n_in; (void)out_size; (void)ws_size;

    // Workspace layout (floats): canvas | series | eidx(int)
    float* canvas = (float*)d_ws;
    float* series = canvas + CANVAS_ELEMS;
    int*   eidx   = (int*)(series + (size_t)NE * 64);

    psf_zero_kernel<<<(CANVAS_ELEMS + 255) / 256, 256, 0, stream>>>(canvas, CANVAS_ELEMS);
    psf_pre_kernel<<<(NE + 255) / 256, 256, 0, stream>>>(xyz, ivs, pc, series, eidx);
    psf_main_kernel<<<NE, 32, 0, stream>>>(coeffs, nph, series, eidx, canvas);
    psf_crop_kernel<<<(OUT_ELEMS + 255) / 256, 256, 0, stream>>>(canvas, (float*)d_out);
}